// FieldWeightedFactorizationMachine_22445499089612
// MI455X (gfx1250) — compile-verified
//
#include <hip/hip_runtime.h>

// FwFM second-order term on gfx1250 via V_WMMA_F32_16X16X4_F32.
// out[n] = 0.5 * sum_l x[l,n] * ( (S x)[l,n] - S[l,l]*x[l,n] ),  S = (W+W^T)/2
// X viewed as [K=39, N=524288]; S padded to 48x40 (3 M-tiles x 10 K-steps of 4).
//
// LDS x layout is row-pair interleaved: word addr = p*32 + col*2 + (row&1),
// pair p = row/2. This makes every WMMA B operand (rows kb,kb+1 at one column,
// kb even) a single aligned ds_load_b64 into an even VGPR pair.

typedef __attribute__((ext_vector_type(2))) float v2f;
typedef __attribute__((ext_vector_type(4))) float v4f;
typedef __attribute__((ext_vector_type(8))) float v8f;

#define K_FIELDS 39
#define NCOLS    (8192 * 64)   // B * D
#define WPB      8             // waves per block (256 threads, wave32)
#define TPW      4             // 16-column tiles per wave
#define PAIRW    32            // words per row-pair (16 cols * 2 rows)
#define TILE_W   (24 * PAIRW)  // 24 pairs (48 padded rows) = 768 words / tile

__global__ __launch_bounds__(256, 1)
void fwfm_wmma_kernel(const float* __restrict__ x,
                      const float* __restrict__ w,
                      float* __restrict__ out)
{
    __shared__ float s_pad[48 * 40];
    __shared__ float s_diag[48];
    __shared__ __align__(16) float x_tile[WPB][TILE_W];  // wave-private buffers

    const int tid  = threadIdx.x;
    const int lane = tid & 31;
    const int wv   = tid >> 5;
    const int hl   = lane >> 4;   // lane half
    const int l16  = lane & 15;

    // ---- Build S_pad = 0.5*(W + W^T), zero padded to 48x40 ----
    for (int i = tid; i < 48 * 40; i += 256) {
        const int r = i / 40, c = i - r * 40;
        float v = 0.0f;
        if (r < K_FIELDS && c < K_FIELDS)
            v = 0.5f * (w[r * K_FIELDS + c] + w[c * K_FIELDS + r]);
        s_pad[i] = v;
    }
    if (tid < 48)
        s_diag[tid] = (tid < K_FIELDS) ? w[tid * K_FIELDS + tid] : 0.0f;

    float* xt = &x_tile[wv][0];
    for (int i = lane; i < TILE_W; i += 32) xt[i] = 0.0f;  // zero rows >= 39 pad
    __syncthreads();

    // ---- S resident in WMMA A-operand registers (reused for every tile) ----
    // A 16x4 f32 layout: lane%16 = M row; K = kk*4 + 2*(lane/16) + v
    v2f a[3][10];
    #pragma unroll
    for (int m = 0; m < 3; ++m) {
        #pragma unroll
        for (int kk = 0; kk < 10; ++kk) {
            const int row = m * 16 + l16;
            const int k0  = kk * 4 + 2 * hl;
            v2f t;
            t[0] = s_pad[row * 40 + k0];
            t[1] = s_pad[row * 40 + k0 + 1];
            a[m][kk] = t;
        }
    }
    // S diagonal in C-layout order: lane half hl holds rows m*16 + hl*8 + v
    float sdr[24];
    #pragma unroll
    for (int m = 0; m < 3; ++m)
        #pragma unroll
        for (int v = 0; v < 8; ++v)
            sdr[m * 8 + v] = s_diag[m * 16 + hl * 8 + v];

    // Per-iteration staging map: pair p = 4r + lane/8, cols c..c+1 = (lane&7)*2
    const int sp = lane >> 3;        // 0..3
    const int sc = (lane & 7) * 2;   // 0,2,..,14

    const long long gw = (long long)blockIdx.x * WPB + wv;
    const long long nbase = gw * (TPW * 16);

    // ---- Register-stage tile 0 (rows 2p and 2p+1, two columns per lane) ----
    float2 pa[5], pb[5];
    #pragma unroll
    for (int r = 0; r < 5; ++r) {
        const int p    = r * 4 + sp;
        const int row1 = 2 * p + 1;
        pa[r] = *(const float2*)&x[(long long)(2 * p) * NCOLS + nbase + sc];
        if (row1 < K_FIELDS)
            pb[r] = *(const float2*)&x[(long long)row1 * NCOLS + nbase + sc];
        else
            pb[r] = make_float2(0.0f, 0.0f);
    }

    for (int t = 0; t < TPW; ++t) {
        const long long n0 = nbase + (long long)t * 16;

        // ---- Commit staged tile to LDS: one aligned b128 per lane per pass ----
        // (same-wave DS ops are in order; stores follow last tile's final reads)
        #pragma unroll
        for (int r = 0; r < 5; ++r) {
            const int p = r * 4 + sp;
            v4f q;
            q[0] = pa[r].x; q[1] = pb[r].x; q[2] = pa[r].y; q[3] = pb[r].y;
            *(v4f*)&xt[p * PAIRW + sc * 2] = q;
        }
        asm volatile("" ::: "memory");

        // ---- Prefetch next tile into registers (latency hidden by WMMAs) ----
        if (t + 1 < TPW) {
            const long long n1 = n0 + 16;
            #pragma unroll
            for (int r = 0; r < 5; ++r) {
                const int p    = r * 4 + sp;
                const int row1 = 2 * p + 1;
                pa[r] = *(const float2*)&x[(long long)(2 * p) * NCOLS + n1 + sc];
                if (row1 < K_FIELDS)
                    pb[r] = *(const float2*)&x[(long long)row1 * NCOLS + n1 + sc];
                else
                    pb[r] = make_float2(0.0f, 0.0f);
            }
        }

        // ---- Y = S_pad * X_tile : 10 K-steps x 3 M-tiles ----
        v8f cc[3] = {v8f{}, v8f{}, v8f{}};
        #pragma unroll
        for (int kk = 0; kk < 10; ++kk) {
            // B 4x16 f32: VGPR v = K row kk*4 + 2*hl + v, col = lane%16
            const int p = 2 * kk + hl;
            const v2f b = *(const v2f*)&xt[p * PAIRW + l16 * 2];
            cc[0] = __builtin_amdgcn_wmma_f32_16x16x4_f32(false, a[0][kk], false, b,
                                                          (short)0, cc[0], false, false);
            cc[1] = __builtin_amdgcn_wmma_f32_16x16x4_f32(false, a[1][kk], false, b,
                                                          (short)0, cc[1], false, false);
            cc[2] = __builtin_amdgcn_wmma_f32_16x16x4_f32(false, a[2][kk], false, b,
                                                          (short)0, cc[2], false, false);
        }

        // ---- Fused epilogue: acc = sum_l x_l*(y_l - S_ll*x_l) ----
        // C/D layout: VGPR v = row m*16 + hl*8 + v, col = lane%16.
        // Rows hl*8+v (v=0..7) are contiguous -> fetch as 4 b64 pairs per m-tile.
        float acc = 0.0f;
        #pragma unroll
        for (int m = 0; m < 3; ++m) {
            #pragma unroll
            for (int pp = 0; pp < 4; ++pp) {
                const int p = m * 8 + hl * 4 + pp;  // pair of rows 2p,2p+1
                const v2f xp = *(const v2f*)&xt[p * PAIRW + l16 * 2];
                const float x0 = xp[0], x1 = xp[1];
                acc += x0 * (cc[m][2 * pp]     - sdr[m * 8 + 2 * pp]     * x0);
                acc += x1 * (cc[m][2 * pp + 1] - sdr[m * 8 + 2 * pp + 1] * x1);
            }
        }
        // Fold the two lane-halves (complementary rows, same column)
        acc += __shfl_xor(acc, 16, 32);
        if (lane < 16)
            out[n0 + lane] = 0.5f * acc;
    }
}

extern "C" void kernel_launch(void* const* d_in, const int* in_sizes, int n_in,
                              void* d_out, int out_size, void* d_ws, size_t ws_size,
                              hipStream_t stream)
{
    const float* x = (const float*)d_in[0];   // [39, 8192, 64] fp32
    const float* w = (const float*)d_in[1];   // [39, 39] fp32
    float* out = (float*)d_out;               // [8192, 64] fp32

    // 1024 blocks * 8 waves * 4 tiles * 16 cols = 524288 = B*D
    dim3 grid(NCOLS / (WPB * TPW * 16));
    dim3 block(256);
    fwfm_wmma_kernel<<<grid, block, 0, stream>>>(x, w, out);
}